// Loss_28716151341290
// MI455X (gfx1250) — compile-verified
//
#include <hip/hip_runtime.h>
#include <hip/hip_bf16.h>

typedef __attribute__((ext_vector_type(2))) float v2f;
typedef __attribute__((ext_vector_type(8))) float v8f;

#define BS    8
#define NROT  60
#define NPTS  500
#define NPAD  512
#define NTILE (NPAD/16)   // 32
#define HW    6400

// ---------------------------------------------------------------------------
// Main chamfer kernel: one block per (b,r); 4 waves; WMMA f32 16x16x4 tiles.
// Fragments are pre-swizzled in LDS so every fragment fetch is a single
// unconditional ds_load_b64 (no divergent selects). All 32 B fragments are
// register-resident across the n-tile loop. The spare K=3 slot carries
// A:=1 / B:=|t|^2 so the WMMA emits  -2*dot(p,t) + |t|^2  directly.
// ---------------------------------------------------------------------------
__global__ __launch_bounds__(128) void k_main(
    const float* __restrict__ pred_r, const float* __restrict__ pred_c,
    const float* __restrict__ target_r, const float* __restrict__ model_points,
    const unsigned char* __restrict__ symmetric, const float* __restrict__ diameters,
    float* __restrict__ per_anchor)
{
  // frag layout: tile t, lane l -> sXf[t*32 + l]
  //   lanes 0..15  (l=col):    (K0,K1)
  //   lanes 16..31 (l=col+16): (K2,K3)
  __shared__ v2f   sAf[NPAD];      // A: (-2x,-2y) | (-2z, 1)
  __shared__ v2f   sBf[NPAD];      // B: ( x,  y ) | ( z, |t|^2 )
  __shared__ float sA2[NPAD];      // |p|^2 per row (0 for pads)
  __shared__ float sS[4], sN[4];

  const int blk  = blockIdx.x;          // b*60 + r
  const int b    = blk / NROT;
  const int tid  = threadIdx.x;
  const int lane = tid & 31;
  const int wid  = tid >> 5;

  // quaternion -> rotation matrix R (reference: pred_pts[n] = R @ mp[n])
  const float qw = pred_r[blk*4+0];
  const float qx = pred_r[blk*4+1];
  const float qy = pred_r[blk*4+2];
  const float qz = pred_r[blk*4+3];
  const float R00 = 1.f - 2.f*(qy*qy+qz*qz), R01 = 2.f*qx*qy - 2.f*qw*qz, R02 =  2.f*qw*qy + 2.f*qx*qz;
  const float R10 = 2.f*qx*qy + 2.f*qz*qw,  R11 = 1.f - 2.f*(qx*qx+qz*qz), R12 = -2.f*qw*qx + 2.f*qy*qz;
  const float R20 = -2.f*qw*qy + 2.f*qx*qz, R21 = 2.f*qw*qx + 2.f*qy*qz,  R22 = 1.f - 2.f*(qx*qx+qy*qy);

  for (int i = tid; i < NPAD; i += 128) {
    const int t = i >> 4, c = i & 15;
    float bx=0.f, by=0.f, bz=0.f, b2=1e30f;   // pad targets: never win the min
    float px=0.f, py=0.f, pz=0.f, a2=0.f;     // pad rows: masked out of the mean
    if (i < NPTS) {
      bx = target_r[(b*NPTS+i)*3+0];
      by = target_r[(b*NPTS+i)*3+1];
      bz = target_r[(b*NPTS+i)*3+2];
      b2 = bx*bx + by*by + bz*bz;
      const float mx = model_points[(b*NPTS+i)*3+0];
      const float my = model_points[(b*NPTS+i)*3+1];
      const float mz = model_points[(b*NPTS+i)*3+2];
      px = R00*mx + R01*my + R02*mz;
      py = R10*mx + R11*my + R12*mz;
      pz = R20*mx + R21*my + R22*mz;
      a2 = px*px + py*py + pz*pz;
    }
    v2f alo = {-2.f*px, -2.f*py};  v2f ahi = {-2.f*pz, 1.f};
    v2f blo = {bx, by};            v2f bhi = {bz, b2};
    sAf[t*32 + c]      = alo;
    sAf[t*32 + c + 16] = ahi;
    sBf[t*32 + c]      = blo;
    sBf[t*32 + c + 16] = bhi;
    sA2[i] = a2;
  }
  __syncthreads();

  const int row = lane & 15;
  const int hi  = lane >> 4;

  // register-resident B fragments (one conflict-free b64 load each)
  v2f bf[NTILE];
  #pragma unroll
  for (int mt = 0; mt < NTILE; ++mt) bf[mt] = sBf[mt*32 + lane];

  float symsum = 0.f;
  // each wave owns n-tiles {wid, wid+4, ...} -> uniform per-wave control flow
  for (int nt = wid; nt < NTILE; nt += 4) {
    const v2f a = sAf[nt*32 + lane];
    float a2r[8];
    #pragma unroll
    for (int g = 0; g < 8; ++g) a2r[g] = sA2[nt*16 + hi*8 + g];
    float mind[8];
    #pragma unroll
    for (int g = 0; g < 8; ++g) mind[g] = 3.0e38f;

    #pragma unroll
    for (int mt = 0; mt < NTILE; ++mt) {
      v8f c = {0.f,0.f,0.f,0.f,0.f,0.f,0.f,0.f};
      // D = A(16x4) x B(4x16) = -2*dot(p,t) + |t|^2  per (n,m)
      c = __builtin_amdgcn_wmma_f32_16x16x4_f32(false, a, false, bf[mt], (short)0, c, false, false);
      #pragma unroll
      for (int g = 0; g < 8; ++g)
        mind[g] = fminf(mind[g], a2r[g] + c[g]);
    }
    // row-min across the 16 lanes of each half-wave, then sqrt + accumulate
    #pragma unroll
    for (int g = 0; g < 8; ++g) {
      float v = mind[g];
      v = fminf(v, __shfl_xor(v, 8, 32));
      v = fminf(v, __shfl_xor(v, 4, 32));
      v = fminf(v, __shfl_xor(v, 2, 32));
      v = fminf(v, __shfl_xor(v, 1, 32));
      const int nrow = nt*16 + hi*8 + g;
      symsum += (row == 0 && nrow < NPTS) ? sqrtf(fmaxf(v, 1e-12f)) : 0.f;
    }
  }

  // non-symmetric branch: mean_n ||pred_pts[n] - tgt[n]|| (coords recovered from frags)
  float nonsum = 0.f;
  for (int n = tid; n < NPTS; n += 128) {
    const int t = n >> 4, c = n & 15;
    const v2f axy = sAf[t*32 + c], az = sAf[t*32 + c + 16];
    const v2f bxy = sBf[t*32 + c], bzv = sBf[t*32 + c + 16];
    const float dx = -0.5f*axy.x - bxy.x;
    const float dy = -0.5f*axy.y - bxy.y;
    const float dz = -0.5f*az.x  - bzv.x;
    nonsum += sqrtf(dx*dx + dy*dy + dz*dz);
  }

  // wave-level fixed-order reductions
  for (int off = 16; off >= 1; off >>= 1) {
    symsum += __shfl_xor(symsum, off, 32);
    nonsum += __shfl_xor(nonsum, off, 32);
  }
  if (lane == 0) { sS[wid] = symsum; sN[wid] = nonsum; }
  __syncthreads();
  if (tid == 0) {
    const float sym_d    = (sS[0]+sS[1]+sS[2]+sS[3]) / (float)NPTS;
    const float nonsym_d = (sN[0]+sN[1]+sN[2]+sN[3]) / (float)NPTS;
    const float d  = symmetric[b] ? sym_d : nonsym_d;
    const float cc = pred_c[blk];
    per_anchor[blk] = d / (diameters[b]*cc) + logf(cc);
  }
}

// ---------------------------------------------------------------------------
// Regularizer: per (b,r) -> where(max_a cos - diag > 1e-3, ., 0)
// ---------------------------------------------------------------------------
__global__ __launch_bounds__(256) void k_reg(
    const float* __restrict__ pred_r, const float* __restrict__ rot_anchors,
    float* __restrict__ reg_out)
{
  const int t = blockIdx.x*blockDim.x + threadIdx.x;
  if (t >= BS*NROT) return;
  const int r = t % NROT;
  const float q0 = pred_r[t*4+0], q1 = pred_r[t*4+1], q2 = pred_r[t*4+2], q3 = pred_r[t*4+3];
  float mx = -3.0e38f, dg = 0.f;
  for (int a = 0; a < NROT; ++a) {
    const float c = q0*rot_anchors[a*4+0] + q1*rot_anchors[a*4+1]
                  + q2*rot_anchors[a*4+2] + q3*rot_anchors[a*4+3];
    mx = fmaxf(mx, c);
    if (a == r) dg = c;
  }
  const float reg = mx - dg;
  reg_out[t] = (reg > 0.001f) ? reg : 0.f;
}

// ---------------------------------------------------------------------------
// Smooth-L1 translation loss: per-block deterministic tree partials
// ---------------------------------------------------------------------------
__global__ __launch_bounds__(256) void k_smooth(
    const float* __restrict__ pred_t, const float* __restrict__ target_t,
    const int* __restrict__ choose, float* __restrict__ partial)
{
  __shared__ float red[256];
  const int idx = blockIdx.x*256 + threadIdx.x;
  float v = 0.f;
  if (idx < BS*NPTS*3) {
    const int b   = idx / (NPTS*3);
    const int rem = idx - b*(NPTS*3);
    const int n   = rem / 3;
    const int c   = rem - n*3;
    const float ts   = target_t[(b*3 + c)*HW + choose[b*NPTS + n]];
    const float diff = pred_t[idx] - ts;
    const float ad   = fabsf(diff);
    v = (ad < 1.f) ? 0.5f*diff*diff : ad - 0.5f;
  }
  red[threadIdx.x] = v;
  __syncthreads();
  for (int s = 128; s > 0; s >>= 1) {
    if (threadIdx.x < s) red[threadIdx.x] += red[threadIdx.x + s];
    __syncthreads();
  }
  if (threadIdx.x == 0) partial[blockIdx.x] = red[0];
}

// ---------------------------------------------------------------------------
// Final deterministic reduction + loss assembly
// ---------------------------------------------------------------------------
__global__ __launch_bounds__(256) void k_final(
    const float* __restrict__ per_anchor, const float* __restrict__ reg_arr,
    const float* __restrict__ tpart, int ntpart, float* __restrict__ out)
{
  __shared__ float red[256];
  __shared__ float res[2];
  const int t = threadIdx.x;

  float v = per_anchor[t];
  if (t + 256 < BS*NROT) v += per_anchor[t + 256];
  red[t] = v; __syncthreads();
  for (int s = 128; s > 0; s >>= 1) { if (t < s) red[t] += red[t+s]; __syncthreads(); }
  if (t == 0) res[0] = red[0];
  __syncthreads();

  v = reg_arr[t];
  if (t + 256 < BS*NROT) v += reg_arr[t + 256];
  red[t] = v; __syncthreads();
  for (int s = 128; s > 0; s >>= 1) { if (t < s) red[t] += red[t+s]; __syncthreads(); }
  if (t == 0) res[1] = red[0];
  __syncthreads();

  v = (t < ntpart) ? tpart[t] : 0.f;
  red[t] = v; __syncthreads();
  for (int s = 128; s > 0; s >>= 1) { if (t < s) red[t] += red[t+s]; __syncthreads(); }

  if (t == 0) {
    const float loss_r   = res[0] / (float)NROT;           // sum_b mean_r
    const float loss_reg = res[1] / (float)(BS*NROT);      // mean
    const float loss_t   = red[0] / (float)(BS*NPTS*3);    // mean
    out[0] = loss_r + 2.f*loss_reg + 5.f*loss_t;
    out[1] = loss_r;
    out[2] = loss_reg;
    out[3] = loss_t;
  }
}

extern "C" void kernel_launch(void* const* d_in, const int* in_sizes, int n_in,
                              void* d_out, int out_size, void* d_ws, size_t ws_size,
                              hipStream_t stream) {
  const float* pred_t       = (const float*)d_in[0];
  const float* pred_r       = (const float*)d_in[1];
  const float* pred_c       = (const float*)d_in[2];
  const float* target_r     = (const float*)d_in[3];
  const float* target_t     = (const float*)d_in[4];
  const float* model_points = (const float*)d_in[5];
  const int*   choose       = (const int*)d_in[6];
  const unsigned char* symmetric = (const unsigned char*)d_in[7];
  const float* diameters    = (const float*)d_in[8];
  const float* rot_anchors  = (const float*)d_in[9];

  float* out   = (float*)d_out;
  float* ws    = (float*)d_ws;
  float* per   = ws;          // 480 floats
  float* reg   = ws + 480;    // 480 floats
  float* tpart = ws + 960;    // 47 floats

  const int ntelem = BS*NPTS*3;
  const int ntpart = (ntelem + 255) / 256;   // 47

  k_main  <<<BS*NROT, 128, 0, stream>>>(pred_r, pred_c, target_r, model_points,
                                        symmetric, diameters, per);
  k_reg   <<<(BS*NROT + 255)/256, 256, 0, stream>>>(pred_r, rot_anchors, reg);
  k_smooth<<<ntpart, 256, 0, stream>>>(pred_t, target_t, choose, tpart);
  k_final <<<1, 256, 0, stream>>>(per, reg, tpart, ntpart, out);
}